// MVN_elbo_autolatent_sp_36206574306032
// MI455X (gfx1250) — compile-verified
//
#include <hip/hip_runtime.h>
#include <math.h>

// ---- problem constants (from setup_inputs) ----
#define DIMS 8
#define NN   2000     // n  (125*16)
#define PP   200      // p  inducing points
#define KAP  20       // latent GPs
#define MMR  10000    // m  (625*16)
#define NW0  1000     // n0
#define NUG  1e-4f

typedef __attribute__((ext_vector_type(2))) float v2f;
typedef __attribute__((ext_vector_type(8))) float v8f;
typedef __attribute__((ext_vector_type(4))) unsigned int u32x4;
typedef __attribute__((ext_vector_type(4))) int i32x4;
typedef __attribute__((ext_vector_type(8))) int i32x8;

#if defined(__has_builtin)
#if __has_builtin(__builtin_amdgcn_tensor_load_to_lds) && \
    __has_builtin(__builtin_amdgcn_s_wait_tensorcnt)
#define HAVE_TDM 1
#endif
#endif
#ifndef HAVE_TDM
#define HAVE_TDM 0
#endif

__device__ __forceinline__ v8f wmma4(v2f a, v2f b, v8f c) {
  // D = A(16x4,f32) * B(4x16,f32) + C(16x16,f32) -> v_wmma_f32_16x16x4_f32
  return __builtin_amdgcn_wmma_f32_16x16x4_f32(false, a, false, b, (short)0, c,
                                               false, false);
}

__device__ __forceinline__ void load_scales(const float* __restrict__ Lmb, int k,
                                            float* einv, float& amp) {
#pragma unroll
  for (int j = 0; j < DIMS; j++) einv[j] = __expf(-Lmb[k * (DIMS + 1) + j]);
  amp = __expf(Lmb[k * (DIMS + 1) + DIMS]);
}

__device__ __forceinline__ float covpair(const float* __restrict__ a,
                                         const float* __restrict__ b,
                                         const float* __restrict__ einv,
                                         float amp) {
  float C = 1.f, V = 0.f;
#pragma unroll
  for (int j = 0; j < DIMS; j++) {
    float s = fabsf(a[j] - b[j]) * einv[j];
    C *= (1.f + s);
    V -= s;
  }
  return amp * C * __expf(V);
}

// ---------- kernel matrices ----------
__global__ void k_build_knn(const float* __restrict__ theta,
                            const float* __restrict__ Lmb,
                            float* __restrict__ Knn, int kbase, int nk) {
  long idx = (long)blockIdx.x * blockDim.x + threadIdx.x;
  if (idx >= (long)nk * NN * NN) return;
  int k = kbase + (int)(idx / ((long)NN * NN));
  int r = (int)(idx % ((long)NN * NN));
  int i = r / NN, j = r % NN;
  float einv[DIMS], amp;
  load_scales(Lmb, k, einv, amp);
  float v = covpair(theta + i * DIMS, theta + j * DIMS, einv, amp);
  if (i == j) v += NUG;
  Knn[idx] = v;
}

__global__ void k_build_cmm(const float* __restrict__ thetai,
                            const float* __restrict__ Lmb,
                            float* __restrict__ Cmm, float* __restrict__ Lm) {
  int idx = blockIdx.x * blockDim.x + threadIdx.x;
  if (idx >= KAP * PP * PP) return;
  int k = idx / (PP * PP), r = idx % (PP * PP);
  int i = r / PP, j = r % PP;
  float einv[DIMS], amp;
  load_scales(Lmb, k, einv, amp);
  float v = covpair(thetai + i * DIMS, thetai + j * DIMS, einv, amp);
  if (i == j) v += NUG;
  Cmm[idx] = v;
  Lm[idx]  = v;   // second copy gets factored in place
}

__global__ void k_build_cnm(const float* __restrict__ theta,
                            const float* __restrict__ thetai,
                            const float* __restrict__ Lmb,
                            float* __restrict__ Cnm) {
  int idx = blockIdx.x * blockDim.x + threadIdx.x;
  if (idx >= KAP * NN * PP) return;
  int k = idx / (NN * PP), r = idx % (NN * PP);
  int i = r / PP, c = r % PP;
  float einv[DIMS], amp;
  load_scales(Lmb, k, einv, amp);
  Cnm[idx] = covpair(theta + i * DIMS, thetai + c * DIMS, einv, amp);
}

// ---------- small batched Cholesky (p=200), right-looking rank-1, in place ----------
__global__ void k_chol_small(float* __restrict__ Abase, int p) {
  float* A = Abase + (size_t)blockIdx.x * p * p;
  for (int j = 0; j < p; j++) {
    if (threadIdx.x == 0) A[j * p + j] = sqrtf(fmaxf(A[j * p + j], 1e-20f));
    __threadfence();
    __syncthreads();
    float d = A[j * p + j];
    for (int i = j + 1 + threadIdx.x; i < p; i += blockDim.x) A[i * p + j] /= d;
    __threadfence();
    __syncthreads();
    int rem = p - j - 1;
    for (int t = threadIdx.x; t < rem * rem; t += blockDim.x) {
      int r = t / rem, c = t % rem;
      if (c <= r) {
        int ii = j + 1 + r, ll = j + 1 + c;
        A[ii * p + ll] -= A[ii * p + j] * A[ll * p + j];
      }
    }
    __threadfence();
    __syncthreads();
  }
}

// ---------- big blocked Cholesky (n=2000, NB=16), one WG per batch entry ----------
// SYRK trailing update runs on v_wmma_f32_16x16x4_f32 with panel chunks staged
// in LDS (stride 17 to avoid bank conflicts).
#define CH 448            // panel chunk rows (28 tiles), 2 bufs => 62 KB LDS
__global__ void k_chol_big(float* __restrict__ Abase) {
  __shared__ float Ld[16][17];
  __shared__ float PA[CH * 17];
  __shared__ float PB[CH * 17];
  float* A = Abase + (size_t)blockIdx.x * NN * NN;
  const int n = NN;
  int tid = threadIdx.x;
  int wave = tid >> 5, lane = tid & 31, hi = lane >> 4, ml = lane & 15;
  for (int s = 0; s < n / 16; s++) {
    int j0 = s * 16;
    if (tid < 256) {
      int r = tid >> 4, c = tid & 15;
      Ld[r][c] = A[(j0 + r) * n + j0 + c];
    }
    __syncthreads();
    if (tid == 0) {  // factor 16x16 diagonal block sequentially
      for (int c = 0; c < 16; c++) {
        float d = Ld[c][c];
        for (int t = 0; t < c; t++) d -= Ld[c][t] * Ld[c][t];
        d = sqrtf(fmaxf(d, 1e-20f));
        Ld[c][c] = d;
        for (int r = c + 1; r < 16; r++) {
          float v = Ld[r][c];
          for (int t = 0; t < c; t++) v -= Ld[r][t] * Ld[c][t];
          Ld[r][c] = v / d;
        }
      }
    }
    __syncthreads();
    if (tid < 256) {
      int r = tid >> 4, c = tid & 15;
      if (c <= r) A[(j0 + r) * n + j0 + c] = Ld[r][c];
    }
    // TRSM: one panel row per thread
    for (int i = j0 + 16 + tid; i < n; i += blockDim.x) {
      float v[16];
#pragma unroll
      for (int c = 0; c < 16; c++) v[c] = A[i * n + j0 + c];
#pragma unroll
      for (int c = 0; c < 16; c++) {
        float x = v[c];
        for (int t = 0; t < c; t++) x -= v[t] * Ld[c][t];
        v[c] = x / Ld[c][c];
      }
#pragma unroll
      for (int c = 0; c < 16; c++) A[i * n + j0 + c] = v[c];
    }
    __threadfence();
    __syncthreads();
    // SYRK trailing update in LDS-staged chunks
    int r0 = j0 + 16;
    int nrows = n - r0;
    int C = (nrows + CH - 1) / CH;
    for (int ci = 0; ci < C; ci++) {
      int ri = r0 + ci * CH;
      int rowsI = min(CH, n - ri);
      for (int t = tid; t < rowsI * 16; t += blockDim.x) {
        int r = t >> 4, c = t & 15;
        PA[r * 17 + c] = A[(ri + r) * n + j0 + c];
      }
      __syncthreads();
      for (int cj = 0; cj <= ci; cj++) {
        int rj = r0 + cj * CH;
        int rowsJ = min(CH, n - rj);
        float* PJ = PA;
        if (cj != ci) {
          for (int t = tid; t < rowsJ * 16; t += blockDim.x) {
            int r = t >> 4, c = t & 15;
            PB[r * 17 + c] = A[(rj + r) * n + j0 + c];
          }
          PJ = PB;
        }
        __syncthreads();
        int ntI = rowsI / 16, ntJ = rowsJ / 16;
        int T = (ci == cj) ? ntI * (ntI + 1) / 2 : ntI * ntJ;
        for (int t = wave; t < T; t += (int)(blockDim.x >> 5)) {  // wave-uniform
          int ti, tj;
          if (ci == cj) {
            ti = (int)((sqrtf(8.f * (float)t + 1.f) - 1.f) * 0.5f);
            while ((ti + 1) * (ti + 2) / 2 <= t) ti++;
            while (ti * (ti + 1) / 2 > t) ti--;
            tj = t - ti * (ti + 1) / 2;
          } else {
            ti = t / ntJ;
            tj = t % ntJ;
          }
          v8f acc = {0.f, 0.f, 0.f, 0.f, 0.f, 0.f, 0.f, 0.f};
#pragma unroll
          for (int kk = 0; kk < 16; kk += 4) {
            int kc = kk + hi * 2;
            v2f a, b;
            a.x = PA[(ti * 16 + ml) * 17 + kc];
            a.y = PA[(ti * 16 + ml) * 17 + kc + 1];
            b.x = PJ[(tj * 16 + ml) * 17 + kc];  // B = P_j^T
            b.y = PJ[(tj * 16 + ml) * 17 + kc + 1];
            acc = wmma4(a, b, acc);
          }
          int Ri = ri + ti * 16, Rj = rj + tj * 16;
          float* dst = A + (Ri + hi * 8) * n + Rj + ml;
          if (Ri != Rj) {
            // off-diagonal tile: uniform branch, straight-line store-back
#pragma unroll
            for (int r = 0; r < 8; r++) dst[r * n] -= acc[r];
          } else {
            // diagonal tile: only lower triangle (row >= col)
#pragma unroll
            for (int r = 0; r < 8; r++)
              if ((r + hi * 8) >= ml) dst[r * n] -= acc[r];
          }
        }
        __syncthreads();
      }
    }
    __threadfence();
    __syncthreads();
  }
}

// ---------- batched triangular solve (vector RHS): y=L\g, x=L'\y ----------
__global__ void k_chol_solve_vec(const float* __restrict__ Lbase, long sL,
                                 const float* __restrict__ gbase, long sg,
                                 float* __restrict__ ybase, long sy,
                                 float* __restrict__ xbase, long sx, int n) {
  __shared__ float red[256];
  long b = blockIdx.x;
  const float* L = Lbase + b * sL;
  const float* g = gbase + b * sg;
  float* y = ybase + b * sy;
  float* x = xbase + b * sx;
  int tid = threadIdx.x;
  for (int j = 0; j < n; j++) {
    float part = 0.f;
    for (int t = tid; t < j; t += 256) part += L[j * n + t] * y[t];
    red[tid] = part;
    __syncthreads();
    for (int w = 128; w > 0; w >>= 1) {
      if (tid < w) red[tid] += red[tid + w];
      __syncthreads();
    }
    if (tid == 0) y[j] = (g[j] - red[0]) / L[j * n + j];
    __syncthreads();
  }
  for (int i = n - 1; i >= 0; i--) {
    float part = 0.f;
    for (int t = i + 1 + tid; t < n; t += 256) part += L[t * n + i] * x[t];
    red[tid] = part;
    __syncthreads();
    for (int w = 128; w > 0; w >>= 1) {
      if (tid < w) red[tid] += red[tid + w];
      __syncthreads();
    }
    if (tid == 0) x[i] = (y[i] - red[0]) / L[i * n + i];
    __syncthreads();
  }
}

// ---------- FITC: A-column solve fused with Delta/Dinv/Dg ----------
__global__ void k_trsm_delta(const float* __restrict__ Lm,
                             const float* __restrict__ Cnm,
                             const float* __restrict__ Lmb,
                             const float* __restrict__ Mu,
                             float* __restrict__ Dinv, float* __restrict__ Dg) {
  int idx = blockIdx.x * blockDim.x + threadIdx.x;
  if (idx >= KAP * NN) return;
  int k = idx / NN, i = idx % NN;
  const float* L = Lm + (size_t)k * PP * PP;
  const float* c = Cnm + ((size_t)k * NN + i) * PP;
  float a[PP];
  float ss = 0.f;
  for (int r = 0; r < PP; r++) {
    float v = c[r];
    for (int t = 0; t < r; t++) v -= L[r * PP + t] * a[t];
    v /= L[r * PP + r];
    a[r] = v;
    ss += v * v;
  }
  float amp = __expf(Lmb[k * (DIMS + 1) + DIMS]);
  float dinv = 1.f / (amp - ss + NUG);
  Dinv[idx] = dinv;
  Dg[idx] = dinv * Mu[idx];
}

// ---------- M = Cmm + Cnm^T diag(Dinv) Cnm  (WMMA, K=2000) ----------
#define MT ((PP + 15) / 16)  // 13
__global__ void k_build_M(const float* __restrict__ Cnm,
                          const float* __restrict__ Dinv,
                          const float* __restrict__ Cmm, float* __restrict__ Mb) {
  int gw = (blockIdx.x * blockDim.x + threadIdx.x) >> 5;
  int lane = threadIdx.x & 31;
  if (gw >= KAP * MT * MT) return;  // wave-uniform
  int k = gw / (MT * MT), rem = gw % (MT * MT);
  int i0 = (rem / MT) * 16, j0 = (rem % MT) * 16;
  int hi = lane >> 4, ml = lane & 15;
  int ia = i0 + ml, jb = j0 + ml;
  v8f acc = {0.f, 0.f, 0.f, 0.f, 0.f, 0.f, 0.f, 0.f};
  for (int t = 0; t < NN; t += 4) {
    int kc = t + hi * 2;
    float d0 = Dinv[k * NN + kc], d1 = Dinv[k * NN + kc + 1];
    const float* r0 = Cnm + ((size_t)k * NN + kc) * PP;
    const float* r1 = r0 + PP;
    v2f a, b;
    a.x = (ia < PP) ? r0[ia] * d0 : 0.f;
    a.y = (ia < PP) ? r1[ia] * d1 : 0.f;
    b.x = (jb < PP) ? r0[jb] : 0.f;
    b.y = (jb < PP) ? r1[jb] : 0.f;
    acc = wmma4(a, b, acc);
  }
#pragma unroll
  for (int r = 0; r < 8; r++) {
    int row = i0 + r + hi * 8, col = j0 + ml;
    if (row < PP && col < PP) {
      size_t o = (size_t)k * PP * PP + row * PP + col;
      Mb[o] = Cmm[o] + acc[r];
    }
  }
}

// ---------- t1 = Cnm^T Dg ----------
__global__ void k_t1(const float* __restrict__ Cnm, const float* __restrict__ Dg,
                     float* __restrict__ t1) {
  int idx = blockIdx.x * blockDim.x + threadIdx.x;
  if (idx >= KAP * PP) return;
  int k = idx / PP, r = idx % PP;
  float s = 0.f;
  for (int i = 0; i < NN; i++) s += Cnm[((size_t)k * NN + i) * PP + r] * Dg[k * NN + i];
  t1[idx] = s;
}

// ---------- Cinv_g = Dg - Dinv * (Cnm @ t2) ----------
__global__ void k_cig(const float* __restrict__ Cnm, const float* __restrict__ t2,
                      const float* __restrict__ Dg, const float* __restrict__ Dinv,
                      float* __restrict__ Cig) {
  int idx = blockIdx.x * blockDim.x + threadIdx.x;
  if (idx >= KAP * NN) return;
  int k = idx / NN, i = idx % NN;
  const float* c = Cnm + ((size_t)k * NN + i) * PP;
  float s = 0.f;
  for (int r = 0; r < PP; r++) s += c[r] * t2[k * PP + r];
  Cig[idx] = Dg[idx] - Dinv[idx] * s;
}

// ---------- fused predictive mean rows ----------
__global__ void k_pred(const float* __restrict__ theta0,
                       const float* __restrict__ theta,
                       const float* __restrict__ Lmb,
                       const float* __restrict__ X, float* __restrict__ out) {
  int idx = blockIdx.x * blockDim.x + threadIdx.x;
  if (idx >= KAP * NW0) return;
  int k = idx / NW0, i0 = idx % NW0;
  float einv[DIMS], amp, t0[DIMS];
  load_scales(Lmb, k, einv, amp);
#pragma unroll
  for (int j = 0; j < DIMS; j++) t0[j] = theta0[i0 * DIMS + j];
  float acc = 0.f;
  for (int c = 0; c < NN; c++)
    acc += covpair(t0, theta + c * DIMS, einv, amp) * X[k * NN + c];
  out[idx] = acc;
}

// ---------- fhat = psi + Phi @ ghat (WMMA, K=20) ----------
// Each 256-thread block handles a 16x128 output slab; the 20x128 ghat slab is
// staged into LDS via the Tensor Data Mover when available.
#define TM (MMR / 16)            // 625
#define TN ((NW0 + 15) / 16)     // 63
#define FH_NG ((TN + 7) / 8)     // 8 groups of 8 column tiles
__global__ void k_fhat(const float* __restrict__ Phi, const float* __restrict__ ghat,
                       const float* __restrict__ psi, float* __restrict__ out) {
  __shared__ float Bs[KAP][128];
  int blk = blockIdx.x;
  int tm = blk / FH_NG, tg = blk % FH_NG;
  int c0 = tg * 128;
#if HAVE_TDM
  if (threadIdx.x < 32) {  // wave 0 issues the TDM descriptor (wave-uniform)
    unsigned ldsoff = (unsigned)(size_t)(&Bs[0][0]);
    unsigned long long ga = (unsigned long long)(size_t)(const void*)(ghat + c0);
    u32x4 g0;
    g0.x = 1u;                                          // count=1
    g0.y = ldsoff;                                      // lds_addr
    g0.z = (unsigned)(ga & 0xffffffffu);                // global_addr lo
    g0.w = (unsigned)((ga >> 32) & 0x01ffffffu) | (2u << 30);  // addr hi | type=2
    int td0 = NW0 - c0;                                 // OOB cols -> zero fill
    i32x8 g1;
    g1[0] = 0x00020000;                     // data_size=4B, no multicast/pad
    g1[1] = (td0 & 0xffff) << 16;           // tensor_dim0 lo16
    g1[2] = ((td0 >> 16) & 0xffff) | (KAP << 16);  // dim0 hi16 | tensor_dim1=20
    g1[3] = (128 << 16);                    // tile_dim0 = 128
    g1[4] = KAP;                            // tile_dim1 = 20, tile_dim2 = 0
    g1[5] = NW0;                            // tensor_dim0_stride = 1000
    g1[6] = 0;
    g1[7] = 0;
    i32x4 z4 = {0, 0, 0, 0};
#if __clang_major__ >= 23
    i32x8 z8 = {0, 0, 0, 0, 0, 0, 0, 0};
    __builtin_amdgcn_tensor_load_to_lds(g0, g1, z4, z4, z8, 0);
#else
    __builtin_amdgcn_tensor_load_to_lds(g0, g1, z4, z4, 0);
#endif
    __builtin_amdgcn_s_wait_tensorcnt(0);
  }
  __syncthreads();
#else
  for (int t = threadIdx.x; t < KAP * 128; t += blockDim.x) {
    int r = t >> 7, c = t & 127;
    Bs[r][c] = (c0 + c < NW0) ? ghat[r * NW0 + c0 + c] : 0.f;
  }
  __syncthreads();
#endif
  int wave = threadIdx.x >> 5, lane = threadIdx.x & 31;
  int hi = lane >> 4, ml = lane & 15;
  int tn = tg * 8 + wave;
  if (tn >= TN) return;  // wave-uniform (only last group, wave 7)
  int mi0 = tm * 16;
  int colL = wave * 16 + ml;
  v8f acc = {0.f, 0.f, 0.f, 0.f, 0.f, 0.f, 0.f, 0.f};
#pragma unroll
  for (int t = 0; t < KAP; t += 4) {
    int kc = t + hi * 2;
    v2f a, b;
    a.x = Phi[(mi0 + ml) * KAP + kc];
    a.y = Phi[(mi0 + ml) * KAP + kc + 1];
    b.x = Bs[kc][colL];
    b.y = Bs[kc + 1][colL];
    acc = wmma4(a, b, acc);
  }
  int col = tn * 16 + ml;
#pragma unroll
  for (int r = 0; r < 8; r++) {
    int row = mi0 + r + hi * 8;
    if (col < NW0) out[(long)row * NW0 + col] = psi[row] + acc[r];
  }
}

// ---------- deterministic sp_diff reduction ----------
__global__ void k_spdiff(const float* __restrict__ ghat,
                         const float* __restrict__ ghatsp, float* __restrict__ out) {
  __shared__ float red[256];
  int tid = threadIdx.x;
  float s = 0.f;
  for (int i = tid; i < KAP * NW0; i += 256) {
    float d = ghat[i] - ghatsp[i];
    s += d * d;
  }
  red[tid] = s;
  __syncthreads();
  for (int w = 128; w > 0; w >>= 1) {
    if (tid < w) red[tid] += red[tid + w];
    __syncthreads();
  }
  if (tid == 0) out[0] = red[0];
}

extern "C" void kernel_launch(void* const* d_in, const int* in_sizes, int n_in,
                              void* d_out, int out_size, void* d_ws, size_t ws_size,
                              hipStream_t stream) {
  const float* theta0 = (const float*)d_in[0];
  const float* Lmb    = (const float*)d_in[1];
  const float* Mu     = (const float*)d_in[2];
  const float* psi    = (const float*)d_in[3];
  const float* Phi    = (const float*)d_in[4];
  const float* theta  = (const float*)d_in[5];
  const float* thetai = (const float*)d_in[6];
  float* out = (float*)d_out;

  float* W = (float*)d_ws;
  size_t off = 0;
  auto alloc = [&](size_t nel) { float* p = W + off; off += nel; return p; };
  float* ghat   = alloc((size_t)KAP * NW0);
  float* ghatsp = alloc((size_t)KAP * NW0);
  float* Xg     = alloc((size_t)KAP * NN);
  float* Cig    = alloc((size_t)KAP * NN);
  float* Dinv   = alloc((size_t)KAP * NN);
  float* Dg     = alloc((size_t)KAP * NN);
  float* Cmm    = alloc((size_t)KAP * PP * PP);
  float* Lm     = alloc((size_t)KAP * PP * PP);
  float* Mb     = alloc((size_t)KAP * PP * PP);
  float* Cnm    = alloc((size_t)KAP * NN * PP);
  float* t1     = alloc((size_t)KAP * PP);
  float* t2     = alloc((size_t)KAP * PP);
  float* ys     = alloc((size_t)KAP * PP);
  // batched big-Cholesky path needs kap*n*n extra scratch (~320 MB); take it
  // if ws_size allows, otherwise fall back to serialized per-k factorization.
  size_t need_batched = (off + (size_t)KAP * NN * NN + (size_t)KAP * NN) * sizeof(float);
  bool batched = ws_size >= need_batched;
  float* Knn = alloc(batched ? (size_t)KAP * NN * NN : (size_t)NN * NN);
  float* yv  = alloc(batched ? (size_t)KAP * NN : (size_t)NN);
  (void)n_in; (void)in_sizes; (void)out_size;

  auto nb = [](long n, int b) { return (unsigned)((n + b - 1) / b); };

  // ---- batched FITC small-matrix path ----
  k_build_cmm<<<nb((long)KAP * PP * PP, 256), 256, 0, stream>>>(thetai, Lmb, Cmm, Lm);
  k_chol_small<<<KAP, 256, 0, stream>>>(Lm, PP);
  k_build_cnm<<<nb((long)KAP * NN * PP, 256), 256, 0, stream>>>(theta, thetai, Lmb, Cnm);
  k_trsm_delta<<<nb((long)KAP * NN, 64), 64, 0, stream>>>(Lm, Cnm, Lmb, Mu, Dinv, Dg);
  k_build_M<<<nb((long)KAP * MT * MT * 32, 256), 256, 0, stream>>>(Cnm, Dinv, Cmm, Mb);
  k_chol_small<<<KAP, 256, 0, stream>>>(Mb, PP);
  k_t1<<<nb((long)KAP * PP, 256), 256, 0, stream>>>(Cnm, Dg, t1);
  k_chol_solve_vec<<<KAP, 256, 0, stream>>>(Mb, (long)PP * PP, t1, PP, ys, PP, t2, PP, PP);
  k_cig<<<nb((long)KAP * NN, 256), 256, 0, stream>>>(Cnm, t2, Dg, Dinv, Cig);

  // ---- dense path: 20 independent n=2000 Cholesky factorizations ----
  if (batched) {
    k_build_knn<<<nb((long)KAP * NN * NN, 256), 256, 0, stream>>>(theta, Lmb, Knn, 0, KAP);
    k_chol_big<<<KAP, 1024, 0, stream>>>(Knn);
    k_chol_solve_vec<<<KAP, 256, 0, stream>>>(Knn, (long)NN * NN, Mu, NN, yv, NN, Xg, NN, NN);
  } else {
    for (int k = 0; k < KAP; k++) {
      k_build_knn<<<nb((long)NN * NN, 256), 256, 0, stream>>>(theta, Lmb, Knn, k, 1);
      k_chol_big<<<1, 1024, 0, stream>>>(Knn);
      k_chol_solve_vec<<<1, 256, 0, stream>>>(Knn, 0, Mu + (size_t)k * NN, 0, yv, 0,
                                              Xg + (size_t)k * NN, 0, NN);
    }
  }

  // ---- fused predictive means, final GEMM, diff ----
  k_pred<<<nb((long)KAP * NW0, 256), 256, 0, stream>>>(theta0, theta, Lmb, Xg, ghat);
  k_pred<<<nb((long)KAP * NW0, 256), 256, 0, stream>>>(theta0, theta, Lmb, Cig, ghatsp);
  k_fhat<<<nb((long)TM * FH_NG, 1), 256, 0, stream>>>(Phi, ghat, psi, out);
  k_spdiff<<<1, 256, 0, stream>>>(ghat, ghatsp, out + (long)MMR * NW0);
}